// CointegrationAttentionLayer_11725260718400
// MI455X (gfx1250) — compile-verified
//
#include <hip/hip_runtime.h>

#define BB 32
#define NN 1024
#define FF 128
#define VV 4000

typedef __attribute__((ext_vector_type(2))) float v2f;
typedef __attribute__((ext_vector_type(8))) float v8f;

// Row stride 1028 dwords: LDS bank = (1028*m + k) % 64 = (4m + k) % 64.
// Lanes 0-15 read banks {4m+k, 4m+k+1}, lanes 16-31 read {4m+k+2, 4m+k+3}:
// all 64 dwords land in distinct banks -> conflict-free b64 loads.
#define S_STRIDE 1028

__launch_bounds__(256)
__global__ void coint_attn_kernel(const float* __restrict__ X,   // [B,N,F]
                                  const int*   __restrict__ IDS, // [B,N]
                                  const float* __restrict__ IM,  // [V,V]
                                  const float* __restrict__ AW,  // [V,V]
                                  float* __restrict__ OUT)       // [B,N,F]
{
    __shared__ float S[16 * S_STRIDE];   // softmax-weighted interaction tile
    __shared__ int   rids[16];
    __shared__ float red_max[8];
    __shared__ float red_sum[8];
    __shared__ float inv_sum[16];

    const int tile = blockIdx.x;      // 0..63 : 16-row tile
    const int b    = blockIdx.y;      // 0..31 : batch
    const int i0   = tile * 16;
    const int tid  = threadIdx.x;     // 0..255  (8 waves of 32)
    const int wid  = tid >> 5;
    const int lane = tid & 31;

    const int* ids_b = IDS + b * NN;

    if (tid < 16) rids[tid] = ids_b[i0 + tid];

    // each thread owns 4 contiguous columns -> one b128 id load
    const int j0 = tid * 4;
    const int4 cid = *(const int4*)(ids_b + j0);
    __syncthreads();

    // ---------------- phase 1: gather + softmax-weighted scores ------------
    for (int i = 0; i < 16; ++i) {
        const int g = i0 + i;                       // global row index (diag test)
        const long rowbase = (long)rids[i] * VV;
        const float* awr = AW + rowbase;
        const float* imr = IM + rowbase;

        // one gather per (i,j) per matrix -- the information-theoretic minimum
        float w0 = (j0 + 0 == g) ? 0.f : fabsf(awr[cid.x]);
        float w1 = (j0 + 1 == g) ? 0.f : fabsf(awr[cid.y]);
        float w2 = (j0 + 2 == g) ? 0.f : fabsf(awr[cid.z]);
        float w3 = (j0 + 3 == g) ? 0.f : fabsf(awr[cid.w]);
        float m0 = (j0 + 0 == g) ? 0.f : imr[cid.x];
        float m1 = (j0 + 1 == g) ? 0.f : imr[cid.y];
        float m2 = (j0 + 2 == g) ? 0.f : imr[cid.z];
        float m3 = (j0 + 3 == g) ? 0.f : imr[cid.w];

        // row max (diag contributes w=0, which |aw|>=0 already covers)
        float mx = fmaxf(fmaxf(w0, w1), fmaxf(w2, w3));
        #pragma unroll
        for (int off = 16; off >= 1; off >>= 1)
            mx = fmaxf(mx, __shfl_xor(mx, off, 32));
        if (lane == 0) red_max[wid] = mx;
        __syncthreads();
        float rmax = red_max[0];
        #pragma unroll
        for (int k = 1; k < 8; ++k) rmax = fmaxf(rmax, red_max[k]);

        // exp; diag term stays in the denominator (w_diag = 0), but the
        // stored score is zero there because inter_diag is masked to 0.
        float e0 = __expf(w0 - rmax);
        float e1 = __expf(w1 - rmax);
        float e2 = __expf(w2 - rmax);
        float e3 = __expf(w3 - rmax);
        float* Srow = S + i * S_STRIDE + j0;
        Srow[0] = m0 * e0;
        Srow[1] = m1 * e1;
        Srow[2] = m2 * e2;
        Srow[3] = m3 * e3;

        float s = (e0 + e1) + (e2 + e3);
        #pragma unroll
        for (int off = 16; off >= 1; off >>= 1)
            s += __shfl_xor(s, off, 32);
        if (lane == 0) red_sum[wid] = s;
        __syncthreads();
        float tot = 0.f;
        #pragma unroll
        for (int k = 0; k < 8; ++k) tot += red_sum[k];
        if (tid == 0) inv_sum[i] = 1.0f / tot;
        __syncthreads();   // also fences red_* reuse for the next row
    }

    // ------------- phase 2: Out[16x128] = S[16x1024] @ X_b[1024x128] -------
    // wave 'wid' owns feature tile columns [wid*16, wid*16+16)
    const int f0   = wid * 16;
    const int half = lane >> 4;     // 0: K=k0,k0+1   1: K=k0+2,k0+3
    const int l16  = lane & 15;     // A: matrix row M   B: matrix col N

    const float* Sa = S + l16 * S_STRIDE + 2 * half;                 // A frag base
    const float* Xk = X + (long)b * NN * FF + f0 + l16 + 2 * half * FF; // B frag base

    v8f acc = {};
    #pragma unroll 4
    for (int k0 = 0; k0 < NN; k0 += 4) {
        v2f a = *(const v2f*)(Sa + k0);      // conflict-free ds b64
        v2f bv;
        bv.x = Xk[k0 * FF];
        bv.y = Xk[k0 * FF + FF];
        acc = __builtin_amdgcn_wmma_f32_16x16x4_f32(
                  /*neg_a=*/false, a, /*neg_b=*/false, bv,
                  /*c_mod=*/(short)0, acc, /*reuse_a=*/false, /*reuse_b=*/false);
    }

    // D layout: VGPR v, lane<16 -> M=v ; lane>=16 -> M=v+8. Apply 1/rowsum.
    float* outp = OUT + (long)b * NN * FF + (long)i0 * FF + f0 + l16;
    #pragma unroll
    for (int v = 0; v < 8; ++v) {
        int m = v + 8 * half;
        outp[(long)m * FF] = acc[v] * inv_sum[m];
    }
}

extern "C" void kernel_launch(void* const* d_in, const int* in_sizes, int n_in,
                              void* d_out, int out_size, void* d_ws, size_t ws_size,
                              hipStream_t stream) {
    const float* X  = (const float*)d_in[0];   // stock_features  [32,1024,128] f32
    const int*   ID = (const int*)  d_in[1];   // stock_ids       [32,1024]     i32
    const float* IM = (const float*)d_in[2];   // interaction     [4000,4000]   f32
    const float* AW = (const float*)d_in[3];   // attention       [4000,4000]   f32
    float* OUT = (float*)d_out;                // [32,1024,128] f32

    dim3 grid(NN / 16, BB);   // 64 row-tiles x 32 batches
    coint_attn_kernel<<<grid, dim3(256), 0, stream>>>(X, ID, IM, AW, OUT);
}